// MINDLoss_72456098283720
// MI455X (gfx1250) — compile-verified
//
#include <hip/hip_runtime.h>
#include <hip/hip_bf16.h>

// MIND loss, MI455X (gfx1250, wave32).
//
// Algebraic reductions vs reference:
//  * shifts are multiplied by 512 -> 72 of 80 shifts give an all-zero shifted
//    image, so their D2 == blur(img2^2) (one shared field, weight 72).
//  * blur is linear -> Vimg needs one blur of the summed 4-shift diff^2 field.
//  * 7x7 Gaussian (sigma=2) is separable -> two 1D 7-tap passes, each cast as
//    a 16x16x4 F32 WMMA chain against a banded Gaussian matrix.
// Total work ~1.5 GFLOP over 8 MB (L2 resident) -> latency bound; one wave per
// 16x16 output tile, 22x22 halo in LDS, deterministic two-stage reduction.

typedef __attribute__((ext_vector_type(2))) float v2f;
typedef __attribute__((ext_vector_type(8))) float v8f;

#define IMG_N 1024
#define EPSV  1e-5f

// 1D Gaussian, sigma=2, 7 taps: exp(-(d-3)^2/8)/sqrt(8*pi)
__device__ __forceinline__ float wgt(int d) {
  float r = 0.0f;
  if (d == 0 || d == 6)      r = 0.06475880f;
  else if (d == 1 || d == 5) r = 0.12098536f;
  else if (d == 2 || d == 4) r = 0.17603266f;
  else if (d == 3)           r = 0.19947114f;
  return r;
}

__device__ __forceinline__ float ldimg(const float* __restrict__ img, int r, int c) {
  return (r >= 0 && r < IMG_N && c >= 0 && c < IMG_N) ? img[r * IMG_N + c] : 0.0f;
}

__global__ __launch_bounds__(32)
void mind_tile_kernel(const float* __restrict__ img1,
                      const float* __restrict__ img2,
                      double* __restrict__ partial) {
  __shared__ float fld[24][24];   // raw field: rows 0..21 valid, 22..23 zero pad
  __shared__ float mid[16][24];   // after vertical pass: cols 0..21 valid, 22..23 zero pad

  const int lane = threadIdx.x;   // 0..31, single wave
  const int hf   = lane >> 4;     // half-wave select
  const int lm   = lane & 15;
  const int R    = blockIdx.y * 16;
  const int C    = blockIdx.x * 16;

  // Zero the K-padding so 0-weight x garbage can never produce NaN.
  if (lane < 24) { fld[22][lane] = 0.0f; fld[23][lane] = 0.0f; }
  if (lane < 16) { mid[lane][22] = 0.0f; mid[lane][23] = 0.0f; }

  // Per-lane Gaussian fragments. A-layout (16x4 f32): lanes 0-15 hold K =
  // 4k,4k+1; lanes 16-31 hold K = 4k+2,4k+3.  G[i,k]=g1[k-i] and H[k,j]=g1[k-j]
  // produce the *same* per-lane values, so W serves as A (vert) and B (horiz).
  v2f W[6];
#pragma unroll
  for (int k = 0; k < 6; ++k) {
    const int kk0 = 4 * k + 2 * hf;
    W[k].x = wgt(kk0 - lm);
    W[k].y = wgt(kk0 + 1 - lm);
  }
  __syncthreads();

  const int SHX[8] = {-1, -1, -1,  0, 0,  1, 1, 1};
  const int SHY[8] = {-1,  0,  1, -1, 1, -1, 0, 1};

  v8f V = {}, m0 = {}, msum = {}, mx = {};

  for (int f = 0; f < 10; ++f) {
    // ---- stage raw 22x22 field (diff^2, zero outside image) into LDS ----
    for (int idx = lane; idx < 22 * 22; idx += 32) {
      const int rr = idx / 22, cc = idx % 22;
      const int gr = R - 3 + rr, gc = C - 3 + cc;
      float v = 0.0f;
      if (gr >= 0 && gr < IMG_N && gc >= 0 && gc < IMG_N) {
        if (f == 0) {                     // summed 4-shift field for Vimg
          const float a  = img1[gr * IMG_N + gc];
          const float d0 = a - ldimg(img1, gr, gc - 512);
          const float d1 = a - ldimg(img1, gr, gc + 512);
          const float d2 = a - ldimg(img1, gr - 512, gc);
          const float d3 = a - ldimg(img1, gr + 512, gc);
          v = d0 * d0 + d1 * d1 + d2 * d2 + d3 * d3;
        } else if (f == 1) {              // zero-shift field (x72 weight)
          const float b = img2[gr * IMG_N + gc];
          v = b * b;
        } else {                          // one of the 8 near shifts
          const int s = f - 2;
          const float b = img2[gr * IMG_N + gc];
          const float d = b - ldimg(img2, gr + SHY[s] * 512, gc + SHX[s] * 512);
          v = d * d;
        }
      }
      fld[rr][cc] = v;
    }
    __syncthreads();

    // ---- vertical 7-tap pass: Out_v = G(16x24) x fld(24x16) per col block ----
#pragma unroll
    for (int cb = 0; cb < 2; ++cb) {
      const int colbase = cb ? 6 : 0;     // cols 0..15 then 6..21 (overlap ok)
      v8f acc = {};
#pragma unroll
      for (int k = 0; k < 6; ++k) {
        const int kk0 = 4 * k + 2 * hf;
        v2f b;
        b.x = fld[kk0][colbase + lm];
        b.y = fld[kk0 + 1][colbase + lm];
        acc = __builtin_amdgcn_wmma_f32_16x16x4_f32(
            false, W[k], false, b, (short)0, acc, false, false);
      }
      // C/D layout: elem e -> row e + 8*hf, col lm
#pragma unroll
      for (int e = 0; e < 8; ++e)
        mid[e + 8 * hf][colbase + lm] = acc[e];
      __syncthreads();
    }

    // ---- horizontal 7-tap pass: Out = mid(16x24) x H(24x16) ----
    v8f h = {};
#pragma unroll
    for (int k = 0; k < 6; ++k) {
      const int kk0 = 4 * k + 2 * hf;
      v2f a;
      a.x = mid[lm][kk0];
      a.y = mid[lm][kk0 + 1];
      h = __builtin_amdgcn_wmma_f32_16x16x4_f32(
          false, a, false, W[k], (short)0, h, false, false);
    }
    __syncthreads();

    // ---- accumulate MIND statistics per output element ----
    if (f == 0) {
#pragma unroll
      for (int e = 0; e < 8; ++e) V[e] = h[e] * 0.25f + EPSV;
    } else {
      v8f m;
#pragma unroll
      for (int e = 0; e < 8; ++e) m[e] = expf(-h[e] / V[e]);
      if (f == 1) {
        m0 = m; mx = m;
#pragma unroll
        for (int e = 0; e < 8; ++e) msum[e] = 0.0f;
      } else {
#pragma unroll
        for (int e = 0; e < 8; ++e) {
          msum[e] += m[e];
          mx[e] = fmaxf(mx[e], m[e]);
        }
      }
    }
    __syncthreads();
  }

  // ---- crop (rows 7..1017, cols 7..1016), per-pixel sum over 80 shifts ----
  float lsum = 0.0f;
#pragma unroll
  for (int e = 0; e < 8; ++e) {
    const int row = R + e + 8 * hf;
    const int col = C + lm;
    if (row >= 7 && row <= 1017 && col >= 7 && col <= 1016)
      lsum += (72.0f * m0[e] + msum[e]) / mx[e];
  }
  // wave32 reduction
#pragma unroll
  for (int off = 16; off > 0; off >>= 1)
    lsum += __shfl_xor(lsum, off, 32);
  if (lane == 0)
    partial[blockIdx.y * gridDim.x + blockIdx.x] = (double)lsum;
}

__global__ void mind_finalize(const double* __restrict__ partial,
                              float* __restrict__ out, int n) {
  if (threadIdx.x == 0 && blockIdx.x == 0) {
    double s = 0.0;
    for (int i = 0; i < n; ++i) s += partial[i];   // fixed order: deterministic
    out[0] = (float)(s / (80.0 * 1011.0 * 1010.0));
  }
}

extern "C" void kernel_launch(void* const* d_in, const int* in_sizes, int n_in,
                              void* d_out, int out_size, void* d_ws, size_t ws_size,
                              hipStream_t stream) {
  const float* img1 = (const float*)d_in[0];
  const float* img2 = (const float*)d_in[1];
  float* out = (float*)d_out;
  double* partial = (double*)d_ws;   // 64*64 doubles = 32 KB scratch

  dim3 grid(IMG_N / 16, IMG_N / 16);
  mind_tile_kernel<<<grid, 32, 0, stream>>>(img1, img2, partial);
  mind_finalize<<<1, 32, 0, stream>>>(partial, out, (IMG_N / 16) * (IMG_N / 16));
}